// PointCN_40364102648452
// MI455X (gfx1250) — compile-verified
//
#include <hip/hip_runtime.h>

#define IN_EPS 1e-3f
#define BN_EPS 1e-5f

typedef float v2f __attribute__((ext_vector_type(2)));
typedef float v8f __attribute__((ext_vector_type(8)));

// ---------------------------------------------------------------------------
// Per-(b,c) column reduction over N: sum and sum-of-squares.
// grid = B*C blocks, 256 threads.
// ---------------------------------------------------------------------------
__global__ void colreduce_kernel(const float* __restrict__ src, int Ctot, int coff,
                                 int Cb, int Npts,
                                 float* __restrict__ osum, float* __restrict__ ossq)
{
    const int bc = blockIdx.x;
    const int b  = bc / Cb;
    const int c  = bc - b * Cb;
    const float* p = src + (size_t)b * Ctot * Npts + (size_t)(coff + c) * Npts;
    float s = 0.f, q = 0.f;
    for (int i = threadIdx.x; i < Npts; i += blockDim.x) {
        float v = p[i];
        s += v;
        q = fmaf(v, v, q);
    }
    __shared__ float ls[256], lq[256];
    ls[threadIdx.x] = s; lq[threadIdx.x] = q;
    __syncthreads();
    for (int st = 128; st > 0; st >>= 1) {
        if (threadIdx.x < st) {
            ls[threadIdx.x] += ls[threadIdx.x + st];
            lq[threadIdx.x] += lq[threadIdx.x + st];
        }
        __syncthreads();
    }
    if (threadIdx.x == 0) { osum[bc] = ls[0]; ossq[bc] = lq[0]; }
}

// ---------------------------------------------------------------------------
// Finalize InstanceNorm+BatchNorm scales. IN output has zero mean and second
// moment v/(v+IN_EPS) per (b,c), so batch mean == 0 and batch var is the
// average of v/(v+IN_EPS) over b. One block, C threads.
//   y = a_bc * x + o_bc  (pre-ReLU, pre-conv)
// ---------------------------------------------------------------------------
__global__ void finalize_in_kernel(const float* __restrict__ sum, const float* __restrict__ ssq,
                                   const float* __restrict__ g, const float* __restrict__ beta,
                                   int Bn, int Cb, int Npts,
                                   float* __restrict__ a, float* __restrict__ o)
{
    const int c = threadIdx.x;
    if (c >= Cb) return;
    const float inv = 1.f / (float)Npts;
    float vb = 0.f;
    for (int b = 0; b < Bn; ++b) {
        float m = sum[b * Cb + c] * inv;
        float v = ssq[b * Cb + c] * inv - m * m;
        vb += v / (v + IN_EPS);
    }
    const float varB = vb / (float)Bn;
    const float s = g[c] * rsqrtf(varB + BN_EPS);
    for (int b = 0; b < Bn; ++b) {
        float m = sum[b * Cb + c] * inv;
        float v = ssq[b * Cb + c] * inv - m * m;
        float ai = s * rsqrtf(v + IN_EPS);
        a[b * Cb + c] = ai;
        o[b * Cb + c] = beta[c] - m * ai;
    }
}

// ---------------------------------------------------------------------------
// Finalize plain per-channel BatchNorm over (B,N). One block, C threads.
// ---------------------------------------------------------------------------
__global__ void finalize_ch_kernel(const float* __restrict__ sum, const float* __restrict__ ssq,
                                   const float* __restrict__ g, const float* __restrict__ beta,
                                   int Bn, int Cb, int Npts,
                                   float* __restrict__ a, float* __restrict__ o)
{
    const int c = threadIdx.x;
    if (c >= Cb) return;
    float S = 0.f, Q = 0.f;
    for (int b = 0; b < Bn; ++b) { S += sum[b * Cb + c]; Q += ssq[b * Cb + c]; }
    const float inv = 1.f / ((float)Bn * (float)Npts);
    const float m = S * inv;
    const float v = Q * inv - m * m;
    const float ai = g[c] * rsqrtf(v + BN_EPS);
    a[c] = ai;
    o[c] = beta[c] - m * ai;
}

// ---------------------------------------------------------------------------
// Fused [optional norm+ReLU] -> 1x1 conv via V_WMMA_F32_16X16X4_F32.
//   MODE 0: y = x
//   MODE 1: y = relu(a[b*Cin+c]*x + o[b*Cin+c])   (per-(b,c) IN+BN)
//   MODE 2: y = relu(a[c]*x + o[c])               (per-channel BN)
// blockIdx.y = batch sample -> all base pointers are block-uniform (SALU),
// per-lane addressing is running 32-bit offsets (SADDR-form loads).
// Each wave owns TWO adjacent 16-point tiles (NT=2): A-frag (W) and norm
// scalars are reused across both, so 2*MT WMMAs per (2*2 + MT + 2) loads.
// A frag: lane l -> M = l&15, K = 2*(l>>4)+v.  C/D: N = l&15, M = v + 8*(l>>4).
// Grid is exact (no partial waves) so EXEC is all 1s, as WMMA requires.
// ---------------------------------------------------------------------------
template <int MT, int MODE>
__global__ void fused_norm_conv_kernel(
    const float* __restrict__ src, int srcCtot, int srcCoff, int Cin,
    const float* __restrict__ Wm, const float* __restrict__ bias,
    const float* __restrict__ a, const float* __restrict__ o,
    float* __restrict__ dst, int dstCtot, int dstCoff, int Npts)
{
    const int lane = threadIdx.x & 31;
    const int wv   = threadIdx.x >> 5;
    const int nl   = lane & 15;
    const int kg   = lane >> 4;
    const int b    = blockIdx.y;
    const int n0   = (blockIdx.x * (int)(blockDim.x >> 5) + wv) * 32 + nl;

    const float* sb = src + (size_t)b * srcCtot * Npts + (size_t)srcCoff * Npts;
    const float* ab = (MODE == 1) ? (a + (size_t)b * Cin) : a;
    const float* ob = (MODE == 1) ? (o + (size_t)b * Cin) : o;

    v8f acc[MT][2];
#pragma unroll
    for (int m = 0; m < MT; ++m) {
        acc[m][0] = v8f{0.f, 0.f, 0.f, 0.f, 0.f, 0.f, 0.f, 0.f};
        acc[m][1] = v8f{0.f, 0.f, 0.f, 0.f, 0.f, 0.f, 0.f, 0.f};
    }

    const unsigned uN = (unsigned)Npts;
    unsigned ox = (unsigned)(2 * kg) * uN + (unsigned)n0;   // x offset, K element 0
    unsigned oa = (unsigned)(2 * kg);                       // a/o offset
    unsigned ow = (unsigned)nl * (unsigned)Cin + (unsigned)(2 * kg);  // W offset (m=0)
    const unsigned xstep = 4u * uN;

    for (int k = 0; k < Cin; k += 4) {
        float x00 = sb[ox];            // K=c0,   tile 0
        float x01 = sb[ox + 16u];      // K=c0,   tile 1
        float x10 = sb[ox + uN];       // K=c0+1, tile 0
        float x11 = sb[ox + uN + 16u]; // K=c0+1, tile 1
        if (MODE != 0) {
            const float a0 = ab[oa], a1 = ab[oa + 1u];
            const float q0 = ob[oa], q1 = ob[oa + 1u];
            x00 = fmaxf(fmaf(a0, x00, q0), 0.f);
            x01 = fmaxf(fmaf(a0, x01, q0), 0.f);
            x10 = fmaxf(fmaf(a1, x10, q1), 0.f);
            x11 = fmaxf(fmaf(a1, x11, q1), 0.f);
        }
        v2f bf0; bf0[0] = x00; bf0[1] = x10;
        v2f bf1; bf1[0] = x01; bf1[1] = x11;
#pragma unroll
        for (int m = 0; m < MT; ++m) {
            const unsigned owm = ow + (unsigned)(m * 16) * (unsigned)Cin;
            v2f af; af[0] = Wm[owm]; af[1] = Wm[owm + 1u];
            acc[m][0] = __builtin_amdgcn_wmma_f32_16x16x4_f32(
                false, af, false, bf0, (short)0, acc[m][0], false, false);
            acc[m][1] = __builtin_amdgcn_wmma_f32_16x16x4_f32(
                false, af, false, bf1, (short)0, acc[m][1], false, false);
        }
        ox += xstep; oa += 4u; ow += 4u;
    }

    float* db = dst + (size_t)b * dstCtot * Npts + (size_t)dstCoff * Npts;
#pragma unroll
    for (int m = 0; m < MT; ++m) {
#pragma unroll
        for (int v = 0; v < 8; ++v) {
            const int co = m * 16 + v + 8 * kg;
            const float bi = bias[co];
            const unsigned od = (unsigned)co * uN + (unsigned)n0;
            db[od]       = acc[m][0][v] + bi;
            db[od + 16u] = acc[m][1][v] + bi;
        }
    }
}

// ---------------------------------------------------------------------------
// AFF global branch: everything on [B=32, C=128] pooled vectors, one block.
// conv(128->32) -> BN(over B) -> ReLU -> conv(32->128) -> BN(over B)
// ---------------------------------------------------------------------------
__global__ void aff_global_kernel(const float* __restrict__ gsum, int Npts,
                                  const float* __restrict__ w1, const float* __restrict__ b1,
                                  const float* __restrict__ g1, const float* __restrict__ bt1,
                                  const float* __restrict__ w2, const float* __restrict__ b2,
                                  const float* __restrict__ g2, const float* __restrict__ bt2,
                                  float* __restrict__ xg)
{
    __shared__ float gm[32 * 128];
    __shared__ float h1[32 * 32];
    __shared__ float h2[32 * 128];
    __shared__ float qa[32], qo[32];
    const int tid = threadIdx.x, ntd = blockDim.x;

    for (int i = tid; i < 32 * 128; i += ntd) gm[i] = gsum[i] / (float)Npts;
    __syncthreads();
    for (int i = tid; i < 32 * 32; i += ntd) {
        const int b = i >> 5, q = i & 31;
        float s = b1[q];
        for (int c = 0; c < 128; ++c) s = fmaf(w1[q * 128 + c], gm[b * 128 + c], s);
        h1[i] = s;
    }
    __syncthreads();
    if (tid < 32) {
        float s = 0.f, ss = 0.f;
        for (int b = 0; b < 32; ++b) { float v = h1[b * 32 + tid]; s += v; ss = fmaf(v, v, ss); }
        float m = s / 32.f, var = ss / 32.f - m * m;
        float ai = g1[tid] * rsqrtf(var + BN_EPS);
        qa[tid] = ai; qo[tid] = bt1[tid] - m * ai;
    }
    __syncthreads();
    for (int i = tid; i < 32 * 32; i += ntd) {
        const int q = i & 31;
        h1[i] = fmaxf(fmaf(qa[q], h1[i], qo[q]), 0.f);
    }
    __syncthreads();
    for (int i = tid; i < 32 * 128; i += ntd) {
        const int b = i >> 7, c = i & 127;
        float s = b2[c];
        for (int q = 0; q < 32; ++q) s = fmaf(w2[c * 32 + q], h1[b * 32 + q], s);
        h2[i] = s;
    }
    __syncthreads();
    if (tid < 128) {
        float s = 0.f, ss = 0.f;
        for (int b = 0; b < 32; ++b) { float v = h2[b * 128 + tid]; s += v; ss = fmaf(v, v, ss); }
        float m = s / 32.f, var = ss / 32.f - m * m;
        float ai = g2[tid] * rsqrtf(var + BN_EPS);
        float oi = bt2[tid] - m * ai;
        for (int b = 0; b < 32; ++b) xg[b * 128 + tid] = fmaf(ai, h2[b * 128 + tid], oi);
    }
}

// ---------------------------------------------------------------------------
// out = sigmoid(bn2(xl_raw) + xg) * init, written in place over xl_raw.
// grid = B*C blocks (one (b,c) plane each).
// ---------------------------------------------------------------------------
__global__ void combine_kernel(float* __restrict__ xl, const float* __restrict__ init,
                               const float* __restrict__ cha, const float* __restrict__ cho,
                               const float* __restrict__ xg, int Npts)
{
    const int bc = blockIdx.x;
    const int c  = bc & 127;
    const float A = cha[c];
    const float O = cho[c] + xg[bc];
    const size_t base = (size_t)bc * Npts;
    for (int i = threadIdx.x; i < Npts; i += blockDim.x) {
        float z = fmaf(A, xl[base + i], O);
        float w = 1.f / (1.f + expf(-z));
        xl[base + i] = w * init[base + i];
    }
}

__global__ void residual_kernel(float* __restrict__ dst, const float* __restrict__ x, size_t total)
{
    size_t i = (size_t)blockIdx.x * blockDim.x + threadIdx.x;
    const size_t stride = (size_t)gridDim.x * blockDim.x;
    for (; i < total; i += stride) dst[i] += x[i];
}

// ---------------------------------------------------------------------------

extern "C" void kernel_launch(void* const* d_in, const int* in_sizes, int n_in,
                              void* d_out, int out_size, void* d_ws, size_t ws_size,
                              hipStream_t stream)
{
    const float* x         = (const float*)d_in[0];
    const float* bn_g_in   = (const float*)d_in[1];   // [2,128]
    const float* bn_b_in   = (const float*)d_in[2];   // [2,128]
    const float* w_in      = (const float*)d_in[3];   // [2,32,128]
    const float* b_in      = (const float*)d_in[4];   // [2,32]
    const float* bn_g_mid  = (const float*)d_in[5];   // [6,32]
    const float* bn_b_mid  = (const float*)d_in[6];   // [6,32]
    const float* w_mid     = (const float*)d_in[7];   // [6,32,32]
    const float* b_mid     = (const float*)d_in[8];   // [6,32]
    const float* aff_w1    = (const float*)d_in[9];   // [2,32,128]
    const float* aff_b1    = (const float*)d_in[10];  // [2,32]
    const float* aff_g1    = (const float*)d_in[11];  // [2,32]
    const float* aff_bt1   = (const float*)d_in[12];  // [2,32]
    const float* aff_w2    = (const float*)d_in[13];  // [2,128,32]
    const float* aff_b2    = (const float*)d_in[14];  // [2,128]
    const float* aff_g2    = (const float*)d_in[15];  // [2,128]
    const float* aff_bt2   = (const float*)d_in[16];  // [2,128]
    float* out             = (float*)d_out;

    const int Bn = 32, Cc = 128;
    const int N  = in_sizes[0] / (Bn * Cc);   // 8000
    const size_t P  = (size_t)Bn * Cc * N;
    const size_t Pq = (size_t)Bn * 32 * N;

    // workspace carve-up (floats)
    float* w = (float*)d_ws;
    float* st_sum = w;  w += 4096;
    float* st_ssq = w;  w += 4096;
    float* aa     = w;  w += 4096;
    float* oo     = w;  w += 4096;
    float* ch_a   = w;  w += 128;
    float* ch_o   = w;  w += 128;
    float* ch_a2  = w;  w += 128;
    float* ch_o2  = w;  w += 128;
    float* gsum   = w;  w += 4096;
    float* gssq   = w;  w += 4096;
    float* xg     = w;  w += 4096;
    float* buf_init = w;        w += P;
    float* buf_xl   = w;        w += P;
    float* buf_h    = w;        w += Pq;

    // conv grid: each wave handles two 16-point tiles; 2 waves per block;
    // grid = (N/64, B) -> exact, every wave full (N=8000 -> 125 blocks/batch)
    const dim3 cgrid(N / 64, Bn);
    const int  cblk = 64;

    // ---- chain 1: i1..i4 into buf_init channels [0..128) ----
    colreduce_kernel<<<Bn * Cc, 256, 0, stream>>>(x, Cc, 0, Cc, N, st_sum, st_ssq);
    finalize_in_kernel<<<1, 128, 0, stream>>>(st_sum, st_ssq, bn_g_in, bn_b_in, Bn, Cc, N, aa, oo);
    fused_norm_conv_kernel<2, 1><<<cgrid, cblk, 0, stream>>>(
        x, Cc, 0, Cc, w_in, b_in, aa, oo, buf_init, Cc, 0, N);
    for (int j = 0; j < 3; ++j) {
        colreduce_kernel<<<Bn * 32, 256, 0, stream>>>(buf_init, Cc, 32 * j, 32, N, st_sum, st_ssq);
        finalize_in_kernel<<<1, 128, 0, stream>>>(st_sum, st_ssq, bn_g_mid + 32 * j, bn_b_mid + 32 * j,
                                                  Bn, 32, N, aa, oo);
        fused_norm_conv_kernel<2, 1><<<cgrid, cblk, 0, stream>>>(
            buf_init, Cc, 32 * j, 32, w_mid + 1024 * j, b_mid + 32 * j, aa, oo,
            buf_init, Cc, 32 * (j + 1), N);
    }

    // ---- AFF local branch ----
    fused_norm_conv_kernel<2, 0><<<cgrid, cblk, 0, stream>>>(
        buf_init, Cc, 0, Cc, aff_w1, aff_b1, nullptr, nullptr, buf_h, 32, 0, N);
    colreduce_kernel<<<Bn * 32, 256, 0, stream>>>(buf_h, 32, 0, 32, N, st_sum, st_ssq);
    finalize_ch_kernel<<<1, 128, 0, stream>>>(st_sum, st_ssq, aff_g1, aff_bt1, Bn, 32, N, ch_a, ch_o);
    fused_norm_conv_kernel<8, 2><<<cgrid, cblk, 0, stream>>>(
        buf_h, 32, 0, 32, aff_w2, aff_b2, ch_a, ch_o, buf_xl, Cc, 0, N);
    colreduce_kernel<<<Bn * Cc, 256, 0, stream>>>(buf_xl, Cc, 0, Cc, N, st_sum, st_ssq);
    finalize_ch_kernel<<<1, 128, 0, stream>>>(st_sum, st_ssq, aff_g2, aff_bt2, Bn, Cc, N, ch_a2, ch_o2);

    // ---- AFF global branch ----
    colreduce_kernel<<<Bn * Cc, 256, 0, stream>>>(buf_init, Cc, 0, Cc, N, gsum, gssq);
    aff_global_kernel<<<1, 256, 0, stream>>>(gsum, N,
        aff_w1 + 4096, aff_b1 + 32, aff_g1 + 32, aff_bt1 + 32,
        aff_w2 + 4096, aff_b2 + 128, aff_g2 + 128, aff_bt2 + 128, xg);

    // ---- gate: buf_xl <- sigmoid(bn2(xl)+xg) * init ----
    combine_kernel<<<Bn * Cc, 256, 0, stream>>>(buf_xl, buf_init, ch_a2, ch_o2, xg, N);

    // ---- chain 2: o1..o4 into d_out channels [0..128) ----
    colreduce_kernel<<<Bn * Cc, 256, 0, stream>>>(buf_xl, Cc, 0, Cc, N, st_sum, st_ssq);
    finalize_in_kernel<<<1, 128, 0, stream>>>(st_sum, st_ssq, bn_g_in + 128, bn_b_in + 128,
                                              Bn, Cc, N, aa, oo);
    fused_norm_conv_kernel<2, 1><<<cgrid, cblk, 0, stream>>>(
        buf_xl, Cc, 0, Cc, w_in + 4096, b_in + 32, aa, oo, out, Cc, 0, N);
    for (int j = 3; j < 6; ++j) {
        const int jj = j - 3;
        colreduce_kernel<<<Bn * 32, 256, 0, stream>>>(out, Cc, 32 * jj, 32, N, st_sum, st_ssq);
        finalize_in_kernel<<<1, 128, 0, stream>>>(st_sum, st_ssq, bn_g_mid + 32 * j, bn_b_mid + 32 * j,
                                                  Bn, 32, N, aa, oo);
        fused_norm_conv_kernel<2, 1><<<cgrid, cblk, 0, stream>>>(
            out, Cc, 32 * jj, 32, w_mid + 1024 * j, b_mid + 32 * j, aa, oo,
            out, Cc, 32 * (jj + 1), N);
    }

    // ---- residual ----
    residual_kernel<<<8192, 256, 0, stream>>>(out, x, P);
}